// HungarianMatcher_84817014161668
// MI455X (gfx1250) — compile-verified
//
#include <hip/hip_runtime.h>
#include <cstdint>
#include <cstddef>

// Problem constants (match reference)
#define BS 128
#define NQ 900
#define NT 128
#define QPB 36                 // queries per block; 900 = 25 * 36
#define BPB (NQ / QPB)         // blocks per batch = 25
#define THREADS 128            // 4 wave32 waves; 4 query-rows in flight, 4 targets/lane

typedef float v4f __attribute__((ext_vector_type(4)));   // native clang vector (b128-capable)

// C[b,q,t] = 5*L1(pred_cxcywh, tgt_xyxy) - 2*GIoU(xyxy(pred), tgt) - sigmoid(logit0)
__global__ __launch_bounds__(THREADS) void matcher_cost_kernel(
    const float* __restrict__ logits,      // [BS, NQ, 2]
    const float* __restrict__ pred_boxes,  // [BS, NQ, 4]  (cx, cy, w, h)
    const float* __restrict__ tgt_boxes,   // [BS, NT, 4]  (x0, y0, x1, y1)
    float* __restrict__ out)               // [BS, NQ, NT]
{
    __shared__ __align__(16) float s_tgt[NT * 4];  // batch's target boxes (2 KB), async-staged
    __shared__ float s_pred[QPB * 4];              // block's pred boxes
    __shared__ float s_parea[QPB];                 // w*h per pred
    __shared__ float s_cls[QPB];                   // -sigmoid(logit0) per pred

    const int tid = threadIdx.x;
    const int b   = blockIdx.x / BPB;
    const int q0  = (blockIdx.x % BPB) * QPB;

    // ---- CDNA5 async copy: global -> LDS, tracked by ASYNCcnt. All 128 lanes, one box each.
    {
        uint32_t lds_off = (uint32_t)(uintptr_t)(&s_tgt[tid * 4]);  // lo32 of flat addr == LDS offset
        const float* gsrc = tgt_boxes + ((size_t)b * NT + tid) * 4;
        asm volatile("global_load_async_to_lds_b128 %0, %1, off"
                     :: "v"(lds_off), "v"(gsrc)
                     : "memory");
    }

    // ---- Stage per-query data (36 queries) with ordinary loads while the async copy flies ----
    if (tid < QPB) {
        const size_t base = (size_t)b * NQ + (q0 + tid);
        v4f p = ((const v4f*)pred_boxes)[base];
        s_pred[tid * 4 + 0] = p.x;
        s_pred[tid * 4 + 1] = p.y;
        s_pred[tid * 4 + 2] = p.z;
        s_pred[tid * 4 + 3] = p.w;
        s_parea[tid] = p.z * p.w;                 // area of xyxy(pred) = w*h
        float l = logits[base * 2];               // class channel 0
        s_cls[tid] = -1.0f / (1.0f + expf(-l));   // cost_class = -sigmoid
    }

    // Drain this wave's outstanding async LDS writes, then block-wide barrier.
    asm volatile("s_wait_asynccnt 0" ::: "memory");
    __syncthreads();

    // ---- Lane owns 4 consecutive targets, kept in registers for all 9 iterations ----
    const int tg    = tid & 31;         // target group 0..31
    const int r     = tid >> 5;         // query-row offset 0..3 (one per wave)
    const int tbase = tg * 4;           // first target column of this lane

    v4f   T[4];
    float A2[4];
    #pragma unroll
    for (int i = 0; i < 4; ++i) {
        T[i]  = ((const v4f*)s_tgt)[tbase + i];              // ds_load_b128
        A2[i] = (T[i].z - T[i].x) * (T[i].w - T[i].y);
    }

    for (int ql = r; ql < QPB; ql += 4) {
        const int q = q0 + ql;
        // LDS broadcast reads (all lanes of a row read the same address -> no conflicts)
        const float cx = s_pred[ql * 4 + 0];
        const float cy = s_pred[ql * 4 + 1];
        const float w  = s_pred[ql * 4 + 2];
        const float h  = s_pred[ql * 4 + 3];
        const float px0 = cx - 0.5f * w, py0 = cy - 0.5f * h;
        const float px1 = cx + 0.5f * w, py1 = cy + 0.5f * h;
        const float area1 = s_parea[ql];
        const float cls   = s_cls[ql];

        v4f res;
        #pragma unroll
        for (int i = 0; i < 4; ++i) {
            const float tx0 = T[i].x, ty0 = T[i].y, tx1 = T[i].z, ty1 = T[i].w;

            // L1 cost: raw pred (cxcywh) vs raw tgt (xyxy), exactly as the reference computes
            const float cb = fabsf(cx - tx0) + fabsf(cy - ty0)
                           + fabsf(w  - tx1) + fabsf(h  - ty1);

            // intersection
            const float ix0 = fmaxf(px0, tx0), iy0 = fmaxf(py0, ty0);
            const float ix1 = fminf(px1, tx1), iy1 = fminf(py1, ty1);
            const float iw = fmaxf(ix1 - ix0, 0.0f), ih = fmaxf(iy1 - iy0, 0.0f);
            const float inter = iw * ih;
            const float uni   = area1 + A2[i] - inter;
            const float iou   = inter / uni;

            // enclosing box
            const float ex0 = fminf(px0, tx0), ey0 = fminf(py0, ty0);
            const float ex1 = fmaxf(px1, tx1), ey1 = fmaxf(py1, ty1);
            const float ew = fmaxf(ex1 - ex0, 0.0f), eh = fmaxf(ey1 - ey0, 0.0f);
            const float ea = ew * eh;

            const float giou = iou - (ea - uni) / ea;
            res[i] = 5.0f * cb - 2.0f * giou + cls;
        }

        // 16B/lane, 512B/wave contiguous non-temporal b128 store of the write-once output
        const size_t idx = ((size_t)b * NQ + q) * NT + tbase;
        __builtin_nontemporal_store(res, (v4f*)&out[idx]);
    }
}

extern "C" void kernel_launch(void* const* d_in, const int* in_sizes, int n_in,
                              void* d_out, int out_size, void* d_ws, size_t ws_size,
                              hipStream_t stream) {
    const float* logits     = (const float*)d_in[0];
    const float* pred_boxes = (const float*)d_in[1];
    const float* tgt_boxes  = (const float*)d_in[2];
    float* out = (float*)d_out;

    dim3 grid(BS * BPB);      // 128 * 25 = 3200 blocks
    dim3 block(THREADS);
    matcher_cost_kernel<<<grid, block, 0, stream>>>(logits, pred_boxes, tgt_boxes, out);
}